// TransformerDecoderAttention_47863115546932
// MI455X (gfx1250) — compile-verified
//
#include <hip/hip_runtime.h>
#include <math.h>

// Sizes fixed by the reference: B=16, M=64, N=256, d=256, H=8, dk=32
#define SCALE_ 0.1767766952966369f  // 1/sqrt(32)

#ifndef __has_builtin
#define __has_builtin(x) 0
#endif

#if defined(__HIP_DEVICE_COMPILE__) && \
    __has_builtin(__builtin_amdgcn_global_load_async_to_lds_b128)
#define USE_ASYNC 1
#else
#define USE_ASYNC 0
#endif

#if USE_ASYNC
#if __has_builtin(__builtin_amdgcn_s_wait_asynccnt)
#define WAIT_ASYNC(N) __builtin_amdgcn_s_wait_asynccnt(N)
#else
#define WAIT_ASYNC(N) asm volatile("s_wait_asynccnt %0" ::"i"(N) : "memory")
#endif
// Types matching the builtin's diagnosed signature:
//   param0: v4i32 __device__* (AS1 global), param1: LDS pointer (AS3)
typedef int v4i_vs __attribute__((vector_size(16)));
typedef __attribute__((address_space(1))) v4i_vs as1_v4i;
typedef __attribute__((address_space(3))) v4i_vs as3_v4i;

// 16B async copy: global -> LDS, tracked by ASYNCcnt.
__device__ __forceinline__ void async_ld16(const float* g, float* lds) {
  __builtin_amdgcn_global_load_async_to_lds_b128(
      (as1_v4i*)(v4i_vs*)const_cast<float*>(g), (as3_v4i*)(v4i_vs*)lds, 0, 0);
}
#endif

typedef float v2f __attribute__((ext_vector_type(2)));
typedef float v8f __attribute__((ext_vector_type(8)));

__device__ __forceinline__ v8f wmma4(v2f a, v2f b, v8f c) {
  // V_WMMA_F32_16X16X4_F32 : D(16x16 f32) = A(16x4) * B(4x16) + C
  return __builtin_amdgcn_wmma_f32_16x16x4_f32(false, a, false, b, (short)0, c,
                                               false, false);
}

__device__ __forceinline__ float mishf(float x) {
  float sp = (x > 20.f) ? x : log1pf(__expf(x));
  return x * tanhf(sp);
}

// ---------------- K0: zero a small buffer ----------------
__global__ void k_zero(float* __restrict__ p, int n) {
  int i = blockIdx.x * blockDim.x + threadIdx.x;
  if (i < n) p[i] = 0.f;
}

// ---------------- K1: prot_e column-sum over all 65536 rows ----------------
__global__ __launch_bounds__(256) void k_prot_e_reduce(
    const float* __restrict__ pe, float* __restrict__ psum) {
  int c = threadIdx.x;
  float s = 0.f;
  size_t base = (size_t)blockIdx.x * 256u * 256u;  // 256 rows per block
  for (int r = 0; r < 256; ++r) s += pe[base + (size_t)r * 256 + c];
  atomicAdd(&psum[c], s);
}

// ---------------- K3: ke_sum, u, beta (single block) ----------------
__global__ __launch_bounds__(256) void k_setup(
    const float* __restrict__ psum, const float* __restrict__ W3,
    const float* __restrict__ b3, const float* __restrict__ W4,
    const float* __restrict__ b4, float* __restrict__ kes,
    float* __restrict__ u, float* __restrict__ beta) {
  __shared__ float sp[256];
  __shared__ float sk[256];
  int t = threadIdx.x;
  sp[t] = psum[t];
  __syncthreads();
  float s = 0.f;
  for (int c = 0; c < 256; ++c) s += sp[c] * W4[t * 256 + c];
  float kv = s + 65536.f * b4[t];  // N*N * b4
  kes[t] = kv; sk[t] = kv;
  __syncthreads();
  for (int h = 0; h < 8; ++h) {
    float a = 0.f;
    for (int j = 0; j < 32; ++j) a += sk[h * 32 + j] * W3[(h * 32 + j) * 256 + t];
    u[h * 256 + t] = a;
  }
  if (t < 8) {
    float a = 0.f;
    for (int j = 0; j < 32; ++j) a += sk[t * 32 + j] * b3[t * 32 + j];
    beta[t] = a;
  }
}

// ---------------- K8: single fused pass over key_edge (67MB) ----------------
// Produces key_edge_sum (sum over n) and edge scores es[b,h,m,n].
// Async double-buffered LDS staging (4 rows / 4KB per issue) when available.
__global__ __launch_bounds__(256) void k_edge_stream(
    const float* __restrict__ ke, const float* __restrict__ u,
    const float* __restrict__ beta, float* __restrict__ kesum,
    float* __restrict__ es) {
  __shared__ float su[2048];
  __shared__ float sb[8];
  __shared__ float part[8][8];  // [wave][head]
#if USE_ASYNC
  __shared__ __align__(16) float ske[2][1024];
#endif
  int t = threadIdx.x, lane = t & 31, w = t >> 5;
  int bm = blockIdx.x, b = bm >> 6, m = bm & 63;
  for (int i = t; i < 2048; i += 256) su[i] = u[i];
  if (t < 8) sb[t] = beta[t];
  __syncthreads();
  float acc = 0.f;
  const float* row = ke + (size_t)bm * 64 * 256;
#if USE_ASYNC
  async_ld16(row + t * 4, &ske[0][t * 4]);
  for (int n4 = 0; n4 < 16; ++n4) {
    int buf = n4 & 1;
    if (n4 + 1 < 16) {
      async_ld16(row + (size_t)(n4 + 1) * 1024 + t * 4, &ske[buf ^ 1][t * 4]);
      WAIT_ASYNC(1);  // newest (just issued) may remain in flight
    } else {
      WAIT_ASYNC(0);
    }
    __syncthreads();  // all waves' async loads for `buf` are complete
    for (int r = 0; r < 4; ++r) {
      int n = n4 * 4 + r;
      float val = ske[buf][r * 256 + t];
#else
  {
    for (int n = 0; n < 64; ++n) {
      float val = row[(size_t)n * 256 + t];
      if (n + 1 < 64) __builtin_prefetch(&row[(size_t)(n + 1) * 256 + t], 0, 1);
#endif
      acc += val;
      float p[8];
#pragma unroll
      for (int h = 0; h < 8; ++h) p[h] = val * su[h * 256 + t];
#pragma unroll
      for (int off = 16; off >= 1; off >>= 1) {
#pragma unroll
        for (int h = 0; h < 8; ++h) p[h] += __shfl_down(p[h], off, 32);
      }
      if (lane == 0) {
#pragma unroll
        for (int h = 0; h < 8; ++h) part[w][h] = p[h];
      }
      __syncthreads();
      if (t < 8) {
        float s = 0.f;
#pragma unroll
        for (int ww = 0; ww < 8; ++ww) s += part[ww][t];
        es[(((size_t)(b * 8 + t) * 64 + m) * 64) + n] = SCALE_ * (s + sb[t]);
      }
      __syncthreads();  // also fences reads of the LDS ring before refill
    }
  }
  kesum[(size_t)bm * 256 + t] = acc;
}

// ---------------- K4: generic C = A @ W^T + bscale*b, d=256, via WMMA ----
// block = 128 thr (4 waves); block tile = 16 rows x 64 cols; A staged in LDS.
// Async path: B128 loads + 16B-chunk XOR swizzle (conflict-free, 16B aligned).
__global__ __launch_bounds__(128) void k_gemm256(
    const float* __restrict__ A, const float* __restrict__ W,
    const float* __restrict__ bias, float bscale, float* __restrict__ C,
    int applyMish) {
#if USE_ASYNC
  __shared__ __align__(16) float sA[16 * 256];
#else
  __shared__ float sA[16 * 257];  // padded stride: bank-conflict free
#endif
  int t = threadIdx.x;
  int row0 = blockIdx.x * 16;
  int colbase = blockIdx.y * 64;
#if USE_ASYNC
#pragma unroll
  for (int i = 0; i < 8; ++i) {
    int chunk = t + i * 128;  // 0..1023 chunks of 4 floats
    int r = chunk >> 6, c4 = chunk & 63;
    async_ld16(A + (size_t)(row0 + r) * 256 + (c4 << 2),
               &sA[r * 256 + ((c4 ^ r) << 2)]);
  }
  WAIT_ASYNC(0);
  __syncthreads();
#else
  for (int i = t; i < 16 * 256; i += 128) {
    int r = i >> 8, c = i & 255;
    sA[r * 257 + c] = A[(size_t)(row0 + r) * 256 + c];
  }
  __syncthreads();
#endif
  int lane = t & 31, w = t >> 5, ln = lane & 15;
  int sel = (lane >> 4) << 1;  // lanes 0-15: K+0/1, lanes 16-31: K+2/3
  int col = colbase + w * 16 + ln;
  const float* wr = W + (size_t)col * 256 + sel;  // B[kk,n] = W[n,kk]
#if !USE_ASYNC
  const float* ar = &sA[ln * 257 + sel];
#endif
  v8f acc = {0, 0, 0, 0, 0, 0, 0, 0};
#pragma unroll 4
  for (int kk = 0; kk < 256; kk += 4) {
    v2f a, bb;
#if USE_ASYNC
    int coff = ((kk >> 2) ^ ln) << 2;
    a.x = sA[ln * 256 + coff + sel];
    a.y = sA[ln * 256 + coff + sel + 1];
#else
    a.x = ar[kk]; a.y = ar[kk + 1];
#endif
    bb.x = wr[kk]; bb.y = wr[kk + 1];
    acc = wmma4(a, bb, acc);
  }
  float bv = bias ? bscale * bias[col] : 0.f;
  int rbase = row0 + ((lane >> 4) << 3);
#pragma unroll
  for (int i = 0; i < 8; ++i) {
    float v = acc[i] + bv;
    if (applyMish) v = mishf(v);
    C[(size_t)(rbase + i) * 256 + col] = v;
  }
}

// ---------------- K5: node scores (per-head q @ k^T, K=32) ----------------
__global__ __launch_bounds__(128) void k_node_scores(
    const float* __restrict__ q, const float* __restrict__ kmat,
    float* __restrict__ sc) {
  int bh = blockIdx.z, b = bh >> 3, h = bh & 7;
  int t = threadIdx.x, lane = t & 31, w = t >> 5, ln = lane & 15;
  int sel = (lane >> 4) << 1;
  int row0 = blockIdx.x * 16;
  int col0 = blockIdx.y * 64 + w * 16;
  const float* qa = q + (size_t)(b * 64 + row0 + ln) * 256 + h * 32 + sel;
  const float* kb = kmat + (size_t)(col0 + ln) * 256 + h * 32 + sel;
  v8f acc = {0, 0, 0, 0, 0, 0, 0, 0};
#pragma unroll
  for (int kk = 0; kk < 32; kk += 4) {
    v2f a, bb;
    a.x = qa[kk]; a.y = qa[kk + 1];
    bb.x = kb[kk]; bb.y = kb[kk + 1];
    acc = wmma4(a, bb, acc);
  }
  int rbase = row0 + ((lane >> 4) << 3);
  int col = col0 + ln;
  float* out = sc + (size_t)bh * 64 * 256;
#pragma unroll
  for (int i = 0; i < 8; ++i)
    out[(size_t)(rbase + i) * 256 + col] = SCALE_ * acc[i];
}

// ---------------- K6a: row softmax of node scores -> message ------------
__global__ __launch_bounds__(256) void k_row_softmax(
    const float* __restrict__ sc, float* __restrict__ msg) {
  __shared__ float red[256];
  int t = threadIdx.x;
  size_t base = (size_t)blockIdx.x * 256;
  float x = sc[base + t];
  red[t] = x; __syncthreads();
  for (int s = 128; s > 0; s >>= 1) {
    if (t < s) red[t] = fmaxf(red[t], red[t + s]);
    __syncthreads();
  }
  float mx = red[0]; __syncthreads();
  float e = __expf(x - mx);
  red[t] = e; __syncthreads();
  for (int s = 128; s > 0; s >>= 1) {
    if (t < s) red[t] += red[t + s];
    __syncthreads();
  }
  msg[base + t] = e / red[0];
}

// ---------------- K6b: column (axis=m) softmax added into message --------
__global__ __launch_bounds__(256) void k_col_softmax_add(
    const float* __restrict__ sc, float* __restrict__ msg) {
  int bh = blockIdx.x, n = threadIdx.x;
  const float* s = sc + (size_t)bh * 64 * 256;
  float* o = msg + (size_t)bh * 64 * 256;
  float mx = -3.4e38f;
  for (int m = 0; m < 64; ++m) mx = fmaxf(mx, s[(size_t)m * 256 + n]);
  float sum = 0.f;
  for (int m = 0; m < 64; ++m) sum += __expf(s[(size_t)m * 256 + n] - mx);
  float inv = 1.f / sum;
  for (int m = 0; m < 64; ++m) {
    size_t idx = (size_t)m * 256 + n;
    o[idx] += __expf(s[idx] - mx) * inv;
  }
}

// ---------------- K6c: msum[b,h,m] = sum_n message ----------------------
__global__ __launch_bounds__(256) void k_msum(const float* __restrict__ msg,
                                              float* __restrict__ msum) {
  __shared__ float red[256];
  int t = threadIdx.x;
  size_t base = (size_t)blockIdx.x * 256;
  red[t] = msg[base + t];
  __syncthreads();
  for (int s = 128; s > 0; s >>= 1) {
    if (t < s) red[t] += red[t + s];
    __syncthreads();
  }
  if (t == 0) msum[blockIdx.x] = red[0];
}

// ---------------- K6d: node_hidden = v * msum ---------------------------
__global__ __launch_bounds__(256) void k_node_hidden(
    const float* __restrict__ v, const float* __restrict__ msum,
    float* __restrict__ nh) {
  int bm = blockIdx.x, c = threadIdx.x;
  int b = bm >> 6, m = bm & 63;
  nh[(size_t)bm * 256 + c] =
      v[(size_t)bm * 256 + c] * msum[(size_t)(b * 8 + (c >> 5)) * 64 + m];
}

// ---------------- K9: edge dual softmax (64x64 tile in LDS) -------------
__global__ __launch_bounds__(64) void k_edge_softmax(
    const float* __restrict__ es, float* __restrict__ Me) {
  __shared__ float tile[4096];
  __shared__ float rmx[64], rsm[64], cmx[64], csm[64];
  int t = threadIdx.x;
  size_t base = (size_t)blockIdx.x * 4096;
  for (int i = t; i < 4096; i += 64) tile[i] = es[base + i];
  __syncthreads();
  float mx = -3.4e38f;
  for (int j = 0; j < 64; ++j) mx = fmaxf(mx, tile[t * 64 + j]);
  float s = 0.f;
  for (int j = 0; j < 64; ++j) s += __expf(tile[t * 64 + j] - mx);
  rmx[t] = mx; rsm[t] = s;
  mx = -3.4e38f;
  for (int j = 0; j < 64; ++j) mx = fmaxf(mx, tile[j * 64 + t]);
  s = 0.f;
  for (int j = 0; j < 64; ++j) s += __expf(tile[j * 64 + t] - mx);
  cmx[t] = mx; csm[t] = s;
  __syncthreads();
  float cm = cmx[t], ci = 1.f / csm[t];
  for (int m = 0; m < 64; ++m) {
    float x = tile[m * 64 + t];
    Me[base + (size_t)m * 64 + t] =
        __expf(x - rmx[m]) / rsm[m] + __expf(x - cm) * ci;
  }
}

// ---------------- K10: G[bm,h,o] = ve_sum_head @ W6_head^T (K=32) -------
__global__ __launch_bounds__(128) void k_G(const float* __restrict__ vesum,
                                           const float* __restrict__ W6,
                                           float* __restrict__ G) {
  int h = blockIdx.z;
  int t = threadIdx.x, lane = t & 31, w = t >> 5, ln = lane & 15;
  int sel = (lane >> 4) << 1;
  int row0 = blockIdx.x * 16;
  int col0 = blockIdx.y * 64 + w * 16;
  const float* aa = vesum + (size_t)(row0 + ln) * 256 + h * 32 + sel;
  const float* wb = W6 + (size_t)(col0 + ln) * 256 + h * 32 + sel;
  v8f acc = {0, 0, 0, 0, 0, 0, 0, 0};
#pragma unroll
  for (int kk = 0; kk < 32; kk += 4) {
    v2f a, bb;
    a.x = aa[kk]; a.y = aa[kk + 1];
    bb.x = wb[kk]; bb.y = wb[kk + 1];
    acc = wmma4(a, bb, acc);
  }
  int rbase = row0 + ((lane >> 4) << 3);
  int col = col0 + ln;
#pragma unroll
  for (int i = 0; i < 8; ++i)
    G[(size_t)(rbase + i) * 2048 + h * 256 + col] = acc[i];
}

// ---------------- K11: edge out = mish(Me @ G + b6), K=8, 67MB write ----
__global__ __launch_bounds__(128) void k_edge_out(
    const float* __restrict__ Me, const float* __restrict__ G,
    const float* __restrict__ b6, float* __restrict__ out0) {
  int bm = blockIdx.x, b = bm >> 6, m = bm & 63;
  int t = threadIdx.x, lane = t & 31, w = t >> 5, ln = lane & 15;
  int sel = (lane >> 4) << 1;
  int col = blockIdx.y * 64 + w * 16 + ln;
  float bv = b6[col];
  const float* gb = G + (size_t)bm * 2048 + col;
  size_t mebase = (size_t)b * 8 * 4096 + (size_t)m * 64;
  int hi = (lane >> 4) << 3;
#pragma unroll
  for (int rt = 0; rt < 4; ++rt) {
    int row0 = rt * 16;
    v8f acc = {0, 0, 0, 0, 0, 0, 0, 0};
#pragma unroll
    for (int kk = 0; kk < 8; kk += 4) {
      int h0 = kk + sel;
      v2f a, bb;
      a.x = Me[mebase + (size_t)h0 * 4096 + row0 + ln];
      a.y = Me[mebase + (size_t)(h0 + 1) * 4096 + row0 + ln];
      bb.x = gb[(size_t)h0 * 256];
      bb.y = gb[(size_t)(h0 + 1) * 256];
      acc = wmma4(a, bb, acc);
    }
#pragma unroll
    for (int i = 0; i < 8; ++i) {
      int n = row0 + hi + i;
      out0[((size_t)bm * 64 + n) * 256 + col] = mishf(acc[i] + bv);
    }
  }
}

extern "C" void kernel_launch(void* const* d_in, const int* in_sizes, int n_in,
                              void* d_out, int out_size, void* d_ws,
                              size_t ws_size, hipStream_t stream) {
  (void)in_sizes; (void)n_in; (void)out_size; (void)ws_size;
  const float* query = (const float*)d_in[0];   // [16,64,256]
  const float* value = (const float*)d_in[1];   // [16,64,256]
  const float* keyE  = (const float*)d_in[2];   // [16,64,64,256]
  const float* protN = (const float*)d_in[4];   // [1,256,256]
  const float* protE = (const float*)d_in[5];   // [1,256,256,256]
  const float* W     = (const float*)d_in[6];   // [8,256,256]
  const float* bias  = (const float*)d_in[7];   // [8,256]

  float* out0 = (float*)d_out;                       // edge  [16,64,64,256]
  float* out1 = out0 + (size_t)16 * 64 * 64 * 256;   // node  [16,64,256]
  float* msg  = out1 + (size_t)16 * 64 * 256;        // message [16,8,64,256]

  float* ws    = (float*)d_ws;
  float* psum  = ws;                  // 256
  float* kes   = ws + 256;            // 256
  float* beta  = ws + 512;            // 8 (padded to 768)
  float* u     = ws + 768;            // 2048
  float* kesum = ws + 2816;           // 262144
  float* qb    = kesum + 262144;      // 262144 (q; later reused as node_hidden)
  float* kb    = qb + 262144;         // 65536
  float* vb    = kb + 65536;          // 262144
  float* vesum = vb + 262144;         // 262144
  float* G     = vesum + 262144;      // 2097152
  float* sc    = G + 2097152;         // 2097152
  float* es    = sc + 2097152;        // 524288
  float* Me    = es + 524288;         // 524288
  float* msum  = Me + 524288;         // 8192   (total ~25.5 MB)

  const float* W0 = W;               const float* b0 = bias;
  const float* W1 = W + 65536;       const float* b1 = bias + 256;
  const float* W2 = W + 2 * 65536;   const float* b2 = bias + 2 * 256;
  const float* W3 = W + 3 * 65536;   const float* b3 = bias + 3 * 256;
  const float* W4 = W + 4 * 65536;   const float* b4 = bias + 4 * 256;
  const float* W5 = W + 5 * 65536;   const float* b5 = bias + 5 * 256;
  const float* W6 = W + 6 * 65536;   const float* b6 = bias + 6 * 256;
  const float* W7 = W + 7 * 65536;   const float* b7 = bias + 7 * 256;

  hipLaunchKernelGGL(k_zero, dim3(1), dim3(256), 0, stream, psum, 256);
  hipLaunchKernelGGL(k_prot_e_reduce, dim3(256), dim3(256), 0, stream, protE, psum);
  hipLaunchKernelGGL(k_setup, dim3(1), dim3(256), 0, stream, psum, W3, b3, W4, b4, kes, u, beta);
  hipLaunchKernelGGL(k_edge_stream, dim3(1024), dim3(256), 0, stream, keyE, u, beta, kesum, es);
  hipLaunchKernelGGL(k_gemm256, dim3(64, 4), dim3(128), 0, stream, query, W0, b0, 1.f, qb, 0);
  hipLaunchKernelGGL(k_gemm256, dim3(16, 4), dim3(128), 0, stream, protN, W1, b1, 1.f, kb, 0);
  hipLaunchKernelGGL(k_gemm256, dim3(64, 4), dim3(128), 0, stream, value, W2, b2, 1.f, vb, 0);
  hipLaunchKernelGGL(k_gemm256, dim3(64, 4), dim3(128), 0, stream, kesum, W5, b5, 64.f, vesum, 0);
  hipLaunchKernelGGL(k_node_scores, dim3(4, 4, 128), dim3(128), 0, stream, qb, kb, sc);
  hipLaunchKernelGGL(k_row_softmax, dim3(8192), dim3(256), 0, stream, sc, msg);
  hipLaunchKernelGGL(k_col_softmax_add, dim3(128), dim3(256), 0, stream, sc, msg);
  hipLaunchKernelGGL(k_msum, dim3(8192), dim3(256), 0, stream, msg, msum);
  hipLaunchKernelGGL(k_node_hidden, dim3(1024), dim3(256), 0, stream, vb, msum, qb);
  hipLaunchKernelGGL(k_gemm256, dim3(64, 4), dim3(128), 0, stream, qb, W7, b7, 1.f, out1, 1);
  hipLaunchKernelGGL(k_edge_softmax, dim3(128), dim3(64), 0, stream, es, Me);
  hipLaunchKernelGGL(k_G, dim3(64, 4, 8), dim3(128), 0, stream, vesum, W6, G);
  hipLaunchKernelGGL(k_edge_out, dim3(1024, 4), dim3(128), 0, stream, Me, G, b6, out0);
}